// multi_shallow_embedding_8641474200290
// MI455X (gfx1250) — compile-verified
//
#include <hip/hip_runtime.h>
#include <stdint.h>

#define G_  16
#define N_  2048
#define K_  16
#define KP1 17

// ---------------------------------------------------------------------------
// Phase A: per-graph selection of the 17 largest and 17 smallest t values.
// 16 blocks (one per graph) x 256 threads. Iterative arg-reduction with
// lowest-index tie-break (matches jax.lax.top_k tie semantics).
// ---------------------------------------------------------------------------
__global__ __launch_bounds__(256) void topk_per_graph(const float* __restrict__ t,
                                                      int* __restrict__ pos_idx,
                                                      int* __restrict__ neg_idx) {
    __shared__ float tvmax[N_];
    __shared__ float tvmin[N_];
    __shared__ float rv[256];
    __shared__ int   ri[256];

    const int g   = blockIdx.x;
    const int tid = threadIdx.x;

    for (int e = tid; e < N_; e += 256) {
        float v = t[(size_t)g * N_ + e];
        tvmax[e] = v;
        tvmin[e] = v;
    }
    __syncthreads();

    // 17 argmax passes (largest t values, rank order)
    for (int k = 0; k < KP1; ++k) {
        float bv = -__builtin_inff();
        int   bi = 0x7fffffff;
        for (int e = tid; e < N_; e += 256) {
            float v = tvmax[e];
            if (v > bv) { bv = v; bi = e; }   // strict > keeps lowest index on ties
        }
        rv[tid] = bv; ri[tid] = bi;
        __syncthreads();
        for (int s = 128; s > 0; s >>= 1) {
            if (tid < s) {
                float v = rv[tid + s]; int j = ri[tid + s];
                if (v > rv[tid] || (v == rv[tid] && j < ri[tid])) { rv[tid] = v; ri[tid] = j; }
            }
            __syncthreads();
        }
        if (tid == 0) { pos_idx[g * KP1 + k] = ri[0]; tvmax[ri[0]] = -__builtin_inff(); }
        __syncthreads();
    }

    // 17 argmin passes (smallest t values, rank order)
    for (int k = 0; k < KP1; ++k) {
        float bv = __builtin_inff();
        int   bi = 0x7fffffff;
        for (int e = tid; e < N_; e += 256) {
            float v = tvmin[e];
            if (v < bv) { bv = v; bi = e; }
        }
        rv[tid] = bv; ri[tid] = bi;
        __syncthreads();
        for (int s = 128; s > 0; s >>= 1) {
            if (tid < s) {
                float v = rv[tid + s]; int j = ri[tid + s];
                if (v < rv[tid] || (v == rv[tid] && j < ri[tid])) { rv[tid] = v; ri[tid] = j; }
            }
            __syncthreads();
        }
        if (tid == 0) { neg_idx[g * KP1 + k] = ri[0]; tvmin[ri[0]] = __builtin_inff(); }
        __syncthreads();
    }
}

// ---------------------------------------------------------------------------
// Phase B: build + stream the 256 MB mask. One block per (g, row).
// Row (8 KB) is composed in LDS, then streamed to HBM with CDNA5 async
// LDS->global stores (ASYNCcnt path) using non-temporal hints (output is
// 256 MB > 192 MB L2, write-once, never re-read by the GPU).
// s_endpgm's implicit wait-idle drains ASYNCcnt before LDS is released.
// ---------------------------------------------------------------------------
__global__ __launch_bounds__(256) void build_mask(const float* __restrict__ s_emb,
                                                  const int* __restrict__ pos_idx,
                                                  const int* __restrict__ neg_idx,
                                                  float* __restrict__ out) {
    __shared__ __align__(16) float row[N_];
    __shared__ int cand[KP1];

    const int bid = blockIdx.x;
    const int g   = bid >> 11;          // N_ == 2048
    const int i   = bid & (N_ - 1);
    const int tid = threadIdx.x;

    // zero the row (512 float4 stores across 256 threads)
    float4  z    = make_float4(0.f, 0.f, 0.f, 0.f);
    float4* row4 = (float4*)row;
    row4[tid]       = z;
    row4[tid + 256] = z;

    // candidate neighbor list by sign of s[g,i]
    const float sv = s_emb[(size_t)g * N_ + i];
    if (tid < KP1) {
        int c;
        if (sv > 0.f)       c = pos_idx[g * KP1 + tid];
        else if (sv < 0.f)  c = neg_idx[g * KP1 + tid];
        else                c = tid;    // all-zero row: top_k tie-breaks to lowest indices
        cand[tid] = c;
    }
    __syncthreads();

    // drop the diagonal candidate (if present) and mark the 16 survivors
    if (tid < K_) {
        int shift = 0;
        for (int p = 0; p <= tid; ++p) shift |= (cand[p] == i) ? 1 : 0;
        row[cand[tid + shift]] = 1.0f;
    }
    __syncthreads();

    // async LDS -> global streaming store of the whole row (non-temporal)
    const float* outrow = out + (size_t)bid * N_;
    uint32_t voff0 = (uint32_t)tid * 16u;        // byte offset within row
    uint32_t voff1 = voff0 + 4096u;
    uint32_t lbase = (uint32_t)(uintptr_t)&row[0];  // low 32 bits of flat ptr = LDS offset
    uint32_t l0 = lbase + voff0;
    uint32_t l1 = lbase + voff1;
    asm volatile("global_store_async_from_lds_b128 %0, %1, %2 th:TH_STORE_NT"
                 :: "v"(voff0), "v"(l0), "s"(outrow) : "memory");
    asm volatile("global_store_async_from_lds_b128 %0, %1, %2 th:TH_STORE_NT"
                 :: "v"(voff1), "v"(l1), "s"(outrow) : "memory");
}

// ---------------------------------------------------------------------------
extern "C" void kernel_launch(void* const* d_in, const int* in_sizes, int n_in,
                              void* d_out, int out_size, void* d_ws, size_t ws_size,
                              hipStream_t stream) {
    (void)in_sizes; (void)n_in; (void)out_size; (void)ws_size;
    const float* emb_s = (const float*)d_in[0];   // (G, N, 1)
    const float* emb_t = (const float*)d_in[1];   // (G, 1, N)
    float*       out   = (float*)d_out;           // (G, N, N)

    int* pos_idx = (int*)d_ws;                    // G * 17 ints
    int* neg_idx = pos_idx + G_ * KP1;            // G * 17 ints

    topk_per_graph<<<G_, 256, 0, stream>>>(emb_t, pos_idx, neg_idx);
    build_mask<<<G_ * N_, 256, 0, stream>>>(emb_s, pos_idx, neg_idx, out);
}